// GNNLayer_6966436954445
// MI455X (gfx1250) — compile-verified
//
#include <hip/hip_runtime.h>

typedef __attribute__((ext_vector_type(2))) float v2f;
typedef __attribute__((ext_vector_type(4))) float v4f;
typedef __attribute__((ext_vector_type(8))) float v8f;

#define N_NODES 768
#define NFEAT   128
#define MFEAT   256
#define GFEAT   256
#define SLOPE   0.01f
#define LN_EPS  1e-5f

// ebar kernel tiling: 8 nodes per block, 32-wide j tiles staged in LDS
#define RPB 8
#define JT  32

__device__ __forceinline__ float lrelu(float x) {
    return fmaxf(x, SLOPE * x);
}

// ---------------------------------------------------------------------------
// Zero the small stat accumulators (S1, S2, gnf[128]) each call.
// ---------------------------------------------------------------------------
__global__ void init_stats_kernel(float* __restrict__ stats /* 2 + NFEAT */) {
    int t = threadIdx.x;
    if (t < 2 + NFEAT) stats[t] = 0.0f;
}

// ---------------------------------------------------------------------------
// Plain fp32 GEMM via V_WMMA_F32_16X16X4_F32. One wave (32 lanes) per 16x16
// output tile. A is MxK row-major, B is KxN row-major, C is MxN row-major.
//
// VGPR layouts (CDNA5 ISA 7.12.2):
//   A (16x4 f32, 2 VGPRs): lanes 0-15 carry M=lane, K={0,1}; lanes 16-31
//     carry M=lane-16, K={2,3}.
//   B (4x16 f32, 2 VGPRs): VGPR0 = rows K=0 (lanes 0-15) / K=2 (16-31),
//     VGPR1 = rows K=1 / K=3, N = lane&15.
//   C/D (16x16 f32, 8 VGPRs): VGPR r = row r (lanes 0-15) / row r+8 (16-31).
// ---------------------------------------------------------------------------
__global__ void wmma_gemm_f32_kernel(const float* __restrict__ A,
                                     const float* __restrict__ B,
                                     float* __restrict__ C,
                                     int M, int K, int N) {
    const int tiles_n = N >> 4;
    const int tile    = blockIdx.x;
    const int tm      = (tile / tiles_n) << 4;
    const int tn      = (tile % tiles_n) << 4;
    const int lane    = threadIdx.x & 31;
    const int half    = lane >> 4;   // selects K pair {0,1} vs {2,3}
    const int l16     = lane & 15;

    v8f acc = {};
    const float* arow = A + (size_t)(tm + l16) * K;
    const float* bcol = B + tn + l16;

    for (int k0 = 0; k0 < K; k0 += 4) {
        const int k = k0 + 2 * half;
        v2f a, b;
        a.x = arow[k];
        a.y = arow[k + 1];
        b.x = bcol[(size_t)k * N];
        b.y = bcol[(size_t)(k + 1) * N];
        acc = __builtin_amdgcn_wmma_f32_16x16x4_f32(
            false, a, false, b, (short)0, acc, false, false);
    }

    float* crow = C + tn + l16;
#pragma unroll
    for (int r = 0; r < 8; ++r) {
        crow[(size_t)(tm + r + 8 * half) * N] = acc[r];
    }
}

// ---------------------------------------------------------------------------
// ebar[i,k] = sum_j lrelu(b[k] + msg[j,k] * edges[i,j]).
//
// VALU-bound elementwise pass (the nonlinearity sits inside the j-sum, so it
// cannot be a matmul). To keep it VALU-bound instead of L2-bound, each block
// handles RPB=8 nodes and stages a JT x MFEAT msg tile (32 KB) in LDS: each
// msg element read from L2 feeds 8 accumulators, cutting L2 traffic 8x
// (~600 MB -> ~75 MB across the grid). Edge values are read as broadcast
// ds_load_b128 pairs (conflict-free).
// ---------------------------------------------------------------------------
__global__ void ebar_kernel(const float* __restrict__ msg,
                            const float* __restrict__ edges,
                            const float* __restrict__ bias,
                            float* __restrict__ ebar) {
    __shared__ float msgT[JT][MFEAT];            // 32 KB
    __shared__ __align__(16) float eT[JT][RPB];  // 1 KB, row = 8 floats

    const int t  = threadIdx.x;          // 0..255
    const int k  = t;                    // message channel owned by thread
    const int i0 = blockIdx.x * RPB;

    const float bk = bias[k];
    float acc[RPB];
#pragma unroll
    for (int r = 0; r < RPB; ++r) acc[r] = 0.0f;

    // cooperative-load index split for the msg tile (float4 per load)
    const int mj = t >> 6;               // 0..3  -> starting j row
    const int mc = (t & 63) << 2;        // 0,4,...,252 -> channel quad

    // edge tile split: row r = t/32, col jj = t%32 (coalesced per row)
    const int er  = t >> 5;              // 0..7
    const int ejj = t & 31;              // 0..31

    for (int j0 = 0; j0 < N_NODES; j0 += JT) {
        // stage msg[j0:j0+JT][0:256] into LDS, 8 x float4 per thread
#pragma unroll
        for (int jj = mj; jj < JT; jj += 4) {
            *(v4f*)&msgT[jj][mc] =
                *(const v4f*)&msg[(size_t)(j0 + jj) * MFEAT + mc];
        }
        // stage edges[i0:i0+8][j0:j0+JT] transposed into LDS
        eT[ejj][er] = edges[(size_t)(i0 + er) * N_NODES + j0 + ejj];
        __syncthreads();

#pragma unroll 4
        for (int jj = 0; jj < JT; ++jj) {
            const float mv = msgT[jj][k];
            const v4f e0 = *(const v4f*)&eT[jj][0];  // broadcast b128
            const v4f e1 = *(const v4f*)&eT[jj][4];  // broadcast b128
            acc[0] += lrelu(fmaf(mv, e0.x, bk));
            acc[1] += lrelu(fmaf(mv, e0.y, bk));
            acc[2] += lrelu(fmaf(mv, e0.z, bk));
            acc[3] += lrelu(fmaf(mv, e0.w, bk));
            acc[4] += lrelu(fmaf(mv, e1.x, bk));
            acc[5] += lrelu(fmaf(mv, e1.y, bk));
            acc[6] += lrelu(fmaf(mv, e1.z, bk));
            acc[7] += lrelu(fmaf(mv, e1.w, bk));
        }
        __syncthreads();
    }

#pragma unroll
    for (int r = 0; r < RPB; ++r)
        ebar[(size_t)(i0 + r) * MFEAT + k] = acc[r];
}

// ---------------------------------------------------------------------------
// nn_pre = lrelu(ebar @ wv) + nodes, fused with accumulation of the global
// first/second moments for the layernorm (wave-level shuffle reduction, then
// one pair of global float atomics per wave).
// ---------------------------------------------------------------------------
__global__ void wmma_gemm_lrelu_add_kernel(const float* __restrict__ A,
                                           const float* __restrict__ B,
                                           const float* __restrict__ addend,
                                           float* __restrict__ C,
                                           float* __restrict__ stats,
                                           int M, int K, int N) {
    const int tiles_n = N >> 4;
    const int tile    = blockIdx.x;
    const int tm      = (tile / tiles_n) << 4;
    const int tn      = (tile % tiles_n) << 4;
    const int lane    = threadIdx.x & 31;
    const int half    = lane >> 4;
    const int l16     = lane & 15;

    v8f acc = {};
    const float* arow = A + (size_t)(tm + l16) * K;
    const float* bcol = B + tn + l16;

    for (int k0 = 0; k0 < K; k0 += 4) {
        const int k = k0 + 2 * half;
        v2f a, b;
        a.x = arow[k];
        a.y = arow[k + 1];
        b.x = bcol[(size_t)k * N];
        b.y = bcol[(size_t)(k + 1) * N];
        acc = __builtin_amdgcn_wmma_f32_16x16x4_f32(
            false, a, false, b, (short)0, acc, false, false);
    }

    float s1 = 0.0f, s2 = 0.0f;
#pragma unroll
    for (int r = 0; r < 8; ++r) {
        const size_t idx = (size_t)(tm + r + 8 * half) * N + tn + l16;
        const float v = lrelu(acc[r]) + addend[idx];
        C[idx] = v;
        s1 += v;
        s2 += v * v;
    }

    // wave32 butterfly reduction
#pragma unroll
    for (int m = 16; m >= 1; m >>= 1) {
        s1 += __shfl_xor(s1, m, 32);
        s2 += __shfl_xor(s2, m, 32);
    }
    if (lane == 0) {
        atomicAdd(&stats[0], s1);
        atomicAdd(&stats[1], s2);
    }
}

// ---------------------------------------------------------------------------
// Layernorm finalize: normalize nn_pre into the output buffer and accumulate
// per-column sums (global node features). blockDim = NFEAT (128), each block
// handles 8 rows; one atomic per (block, column).
// ---------------------------------------------------------------------------
__global__ void layernorm_gnf_kernel(const float* __restrict__ nn_pre,
                                     const float* __restrict__ stats,
                                     float* __restrict__ out_nodes,
                                     float* __restrict__ gnf) {
    const int ROWS = 8;
    const int n  = threadIdx.x;            // column 0..127
    const int r0 = blockIdx.x * ROWS;

    const float inv_cnt = 1.0f / (float)(N_NODES * NFEAT);
    const float mu  = stats[0] * inv_cnt;
    const float var = stats[1] * inv_cnt - mu * mu;
    const float rs  = rsqrtf(var + LN_EPS);

    float csum = 0.0f;
#pragma unroll
    for (int r = 0; r < ROWS; ++r) {
        const size_t idx = (size_t)(r0 + r) * NFEAT + n;
        const float v = (nn_pre[idx] - mu) * rs;
        out_nodes[idx] = v;
        csum += v;
    }
    atomicAdd(&gnf[n], csum);
}

// ---------------------------------------------------------------------------
// new_features = lrelu(gnf @ wu) + features.  128x256 vec-mat: one block of
// 256 threads, each owning one output channel.
// ---------------------------------------------------------------------------
__global__ void feat_kernel(const float* __restrict__ gnf,
                            const float* __restrict__ wu,
                            const float* __restrict__ features,
                            float* __restrict__ out_feat) {
    const int g = threadIdx.x;   // 0..GFEAT-1
    float acc = 0.0f;
#pragma unroll 8
    for (int k = 0; k < NFEAT; ++k)
        acc = fmaf(gnf[k], wu[(size_t)k * GFEAT + g], acc);
    out_feat[g] = lrelu(acc) + features[g];
}

// ---------------------------------------------------------------------------
extern "C" void kernel_launch(void* const* d_in, const int* in_sizes, int n_in,
                              void* d_out, int out_size, void* d_ws, size_t ws_size,
                              hipStream_t stream) {
    const float* nodes    = (const float*)d_in[0];  // [768,128]
    const float* edges    = (const float*)d_in[1];  // [768,768]
    const float* features = (const float*)d_in[2];  // [256]
    const float* we       = (const float*)d_in[3];  // [128,256]
    const float* b        = (const float*)d_in[4];  // [256]
    const float* wv       = (const float*)d_in[5];  // [256,128]
    const float* wu       = (const float*)d_in[6];  // [128,256]

    float* ws     = (float*)d_ws;
    float* msg    = ws;                               // 768*256
    float* ebar   = msg  + (size_t)N_NODES * MFEAT;   // 768*256
    float* nn_pre = ebar + (size_t)N_NODES * MFEAT;   // 768*128
    float* stats  = nn_pre + (size_t)N_NODES * NFEAT; // [S1, S2, gnf[128]]
    float* gnf    = stats + 2;

    float* out_nodes = (float*)d_out;                        // 768*128
    float* out_edges = out_nodes + (size_t)N_NODES * NFEAT;  // 768*768
    float* out_feat  = out_edges + (size_t)N_NODES * N_NODES;// 256

    // 0) zero accumulators
    init_stats_kernel<<<1, 256, 0, stream>>>(stats);

    // 1) msg = nodes @ we   [768x128]@[128x256]
    {
        dim3 grid((N_NODES / 16) * (MFEAT / 16));
        wmma_gemm_f32_kernel<<<grid, 32, 0, stream>>>(
            nodes, we, msg, N_NODES, NFEAT, MFEAT);
    }

    // 2) ebar[i] = sum_j lrelu(b + msg[j] * edges[i,j]), 8 nodes per block
    ebar_kernel<<<N_NODES / RPB, MFEAT, 0, stream>>>(msg, edges, b, ebar);

    // 3) nn_pre = lrelu(ebar @ wv) + nodes, accumulate moments
    {
        dim3 grid((N_NODES / 16) * (NFEAT / 16));
        wmma_gemm_lrelu_add_kernel<<<grid, 32, 0, stream>>>(
            ebar, wv, nodes, nn_pre, stats, N_NODES, MFEAT, NFEAT);
    }

    // 4) layernorm + column sums -> out_nodes, gnf
    layernorm_gnf_kernel<<<N_NODES / 8, NFEAT, 0, stream>>>(
        nn_pre, stats, out_nodes, gnf);

    // 5) new_features = lrelu(gnf @ wu) + features
    feat_kernel<<<1, GFEAT, 0, stream>>>(gnf, wu, features, out_feat);

    // 6) edges pass-through
    hipMemcpyAsync(out_edges, edges,
                   (size_t)N_NODES * N_NODES * sizeof(float),
                   hipMemcpyDeviceToDevice, stream);
}